// OuterProductNetwork_58377195487415
// MI455X (gfx1250) — compile-verified
//
#include <hip/hip_runtime.h>

// OuterProductNetwork: kp[b,n] = sum_{d,e} x[b,row[n],d] * kernel[e,n,d] * x[b,col[n],e]
// Strategy: 1 workgroup per pair n (780 WGs, 256 thr = 8 waves, wave32).
//  - wave w owns e-slice [16w,16w+16); K_n B-fragments live in 64 VGPRs for the
//    whole workgroup (K_n read from HBM exactly once per WG, nontemporal).
//  - per 16-batch tile: P/Q rows staged in LDS (padded stride 132 -> conflict-free
//    ds_load_2addr_b64), 32x v_wmma_f32_16x16x4_f32 accumulate T = P @ K_n^T,
//    then reduce T*Q with half-wave shuffles and a cross-wave LDS combine.
//  - software pipeline: tile t+1 is prefetched from global into registers while
//    tile t's WMMAs execute; ds_store happens at the top of the next iteration.

#define NUM_FIELDS 40
#define EMBED      128
#define BATCH      1024
#define NUM_IX     780
#define TILE_B     16
#define NTILES     (BATCH / TILE_B)
#define LDS_STRIDE 132   // 128 + 4 pad: lane m hits bank 4m+c -> conflict-free b64

typedef __attribute__((ext_vector_type(2))) float v2f;
typedef __attribute__((ext_vector_type(4))) float v4f;
typedef __attribute__((ext_vector_type(8))) float v8f;

__global__ __launch_bounds__(256)
void opnet_pair_kernel(const float* __restrict__ x,
                       const float* __restrict__ kern,
                       float* __restrict__ out) {
  const int n    = blockIdx.x;        // pair index
  const int tid  = threadIdx.x;
  const int wave = tid >> 5;          // 0..7
  const int lane = tid & 31;
  const int lo   = lane & 15;
  const int hi   = lane >> 4;

  // Decode (row, col) field indices from flat upper-triangular pair index n.
  int fi = 0, rem = n, cnt = NUM_FIELDS - 1;
  while (rem >= cnt) { rem -= cnt; ++fi; --cnt; }
  const int fj = fi + 1 + rem;

  __shared__ float Plds[TILE_B * LDS_STRIDE];
  __shared__ float Qlds[TILE_B * LDS_STRIDE];
  __shared__ float Part[8][TILE_B];

  // ---- B fragments: B[K=d][N=e] = K_n[e][d], e = 16*wave + lo ----
  // 16x16x4 f32 B layout: v[0] = row K = 2*hi, v[1] = row K = 2*hi+1, col N = lo.
  // Per k-step the lane needs float2 at K_n[e][4k + 2*hi]; row e is contiguous.
  const int e = 16 * wave + lo;
  const float* krow = kern + (size_t)e * ((size_t)NUM_IX * EMBED)
                           + (size_t)n * EMBED + 2 * hi;
  v2f bfrag[32];
#pragma unroll
  for (int k = 0; k < 32; ++k) {
    bfrag[k] = __builtin_nontemporal_load((const v2f*)(krow + 4 * k));
  }

  // Cooperative staging geometry: 16 threads per batch row, 8 floats per thread.
  const int sr = tid >> 4;            // 0..15 : batch row within tile
  const int sc = (tid & 15) * 8;      // start col (2x float4)
  const float* xp = x + ((size_t)sr * NUM_FIELDS + fi) * EMBED + sc;
  const float* xq = x + ((size_t)sr * NUM_FIELDS + fj) * EMBED + sc;
  const size_t xstep = (size_t)TILE_B * NUM_FIELDS * EMBED;  // advance one tile

  // ---- prologue: prefetch tile 0 into registers ----
  v4f p0 = *(const v4f*)(xp);
  v4f p1 = *(const v4f*)(xp + 4);
  v4f q0 = *(const v4f*)(xq);
  v4f q1 = *(const v4f*)(xq + 4);

  for (int t = 0; t < NTILES; ++t) {
    __syncthreads();   // previous tile's LDS reads (P/Q/Part) complete

    // ---- commit prefetched tile t to LDS ----
    *(v4f*)&Plds[sr * LDS_STRIDE + sc]     = p0;
    *(v4f*)&Plds[sr * LDS_STRIDE + sc + 4] = p1;
    *(v4f*)&Qlds[sr * LDS_STRIDE + sc]     = q0;
    *(v4f*)&Qlds[sr * LDS_STRIDE + sc + 4] = q1;
    __syncthreads();

    // ---- issue prefetch of tile t+1 (covered by the 32 WMMAs below) ----
    if (t + 1 < NTILES) {
      const float* np = xp + (size_t)(t + 1) * xstep;
      const float* nq = xq + (size_t)(t + 1) * xstep;
      p0 = *(const v4f*)(np);
      p1 = *(const v4f*)(np + 4);
      q0 = *(const v4f*)(nq);
      q1 = *(const v4f*)(nq + 4);
    }

    // ---- T = P @ K_n^T over this wave's 16-wide e-slice (K=128 in 32 steps) ----
    // A layout: lane holds P[m = lo][d = 4k + 2*hi + {0,1}] as float2 (row-major LDS).
    v8f acc = {};
#pragma unroll
    for (int k = 0; k < 32; ++k) {
      v2f a = *(const v2f*)&Plds[lo * LDS_STRIDE + 4 * k + 2 * hi];
      acc = __builtin_amdgcn_wmma_f32_16x16x4_f32(
          /*neg_a=*/false, a, /*neg_b=*/false, bfrag[k],
          /*c_mod=*/(short)0, acc, /*reuse_a=*/false, /*reuse_b=*/false);
    }

    // ---- partial[m] = sum_{e in slice} T[m][e] * Q[m][e] ----
    // C/D layout: acc[r] = T[M = r + 8*hi][N = lo]  (global e = 16*wave + lo)
#pragma unroll
    for (int r = 0; r < 8; ++r) {
      const int m = r + 8 * hi;
      float v = acc[r] * Qlds[m * LDS_STRIDE + e];
      v += __shfl_xor(v, 1, 32);   // xor masks < 16: halves reduce independently
      v += __shfl_xor(v, 2, 32);
      v += __shfl_xor(v, 4, 32);
      v += __shfl_xor(v, 8, 32);
      if (lo == 0) Part[wave][m] = v;
    }
    __syncthreads();

    // ---- cross-wave combine + store column n for this batch tile ----
    if (tid < TILE_B) {
      float s = 0.f;
#pragma unroll
      for (int w = 0; w < 8; ++w) s += Part[w][tid];
      out[(size_t)(t * TILE_B + tid) * NUM_IX + n] = s;
    }
  }
}

extern "C" void kernel_launch(void* const* d_in, const int* in_sizes, int n_in,
                              void* d_out, int out_size, void* d_ws, size_t ws_size,
                              hipStream_t stream) {
  const float* x    = (const float*)d_in[0];   // (1024, 40, 128) f32
  const float* kern = (const float*)d_in[1];   // (128, 780, 128) f32
  float* out = (float*)d_out;                  // (1024, 780) f32
  (void)in_sizes; (void)n_in; (void)out_size; (void)d_ws; (void)ws_size;
  opnet_pair_kernel<<<NUM_IX, 256, 0, stream>>>(x, kern, out);
}